// SequenceLoss_29377576305144
// MI455X (gfx1250) — compile-verified
//
#include <hip/hip_runtime.h>

// ---------------------------------------------------------------------------
// SequenceLoss for MI455X (gfx1250, wave32).
// One wave32 per batch element; HUX ring recurrence and the Soft-DTW
// anti-diagonal wavefront are held entirely in VGPRs with cross-lane
// rotates (no barriers on the serial chains). Pearson Gram matrix via
// v_wmma_f32_16x16x4_f32.
// ---------------------------------------------------------------------------

#define TLEN   512
#define KP     16          // elements per lane (TLEN / 32)
#define NRSTEP 200
#define BIGF   1.0e8f
#define GAMMAF 0.1f

typedef __attribute__((ext_vector_type(2))) float v2f;
typedef __attribute__((ext_vector_type(8))) float v8f;

__device__ __forceinline__ float waveRedSum(float v) {
#pragma unroll
  for (int m = 16; m >= 1; m >>= 1) v += __shfl_xor(v, m, 32);
  return v;
}

__global__ __launch_bounds__(32) void hux_dtw_kernel(
    const float* __restrict__ pred_map,   // [B,1,T,W]
    const int*   __restrict__ lat_idx,    // [B,T]
    const float* __restrict__ omega,      // [B]
    const float* __restrict__ omni_ts,    // [B,T]
    float* __restrict__ v_out,            // [B,T]  (front of d_out)
    float* __restrict__ ws,               // [B*4] partials
    int W)
{
  __shared__ float Ylds[TLEN];  // v_out_scaled, random-access j for DTW
  __shared__ float pcL[TLEN];   // centered pred   (for WMMA marshaling)
  __shared__ float tcL[TLEN];   // centered target (for WMMA marshaling)

  const int b = blockIdx.x;
  const int l = threadIdx.x;    // lane 0..31 (single wave per block)

  // constants (match reference: dr = (R1-R0)/NR, dphi = 2*pi/T)
  const float dr   = (float)((215.0 - 21.5) * 695700.0 / 200.0); // 673089.75
  const float dphi = (float)(6.283185307179586476925287 / 512.0);
  const float om   = omega[b];
  const float num  = dr * om;   // dr*om, grouped as in reference

  // ---- gather v_in and load omni (t = k*32 + l interleaved layout) ----
  float v[KP], X[KP];
#pragma unroll
  for (int k = 0; k < KP; ++k) {
    const int t = k * 32 + l;
    const int idx = lat_idx[b * TLEN + t];
    v[k] = pred_map[(size_t)b * TLEN * W + (size_t)t * W + idx];
    X[k] = (omni_ts[b * TLEN + t] - 200.0f) / 1000.0f;
  }

  // ---- HUX-f: 200 serial upwind steps, ring neighbor via lane rotate ----
#pragma unroll 1
  for (int s = 0; s < NRSTEP; ++s) {
    float rot[KP];
#pragma unroll
    for (int k = 0; k < KP; ++k)
      rot[k] = __shfl(v[k], (l + 1) & 31, 32);   // lane l gets lane l+1
#pragma unroll
    for (int k = 0; k < KP; ++k) {
      const float vn = (l == 31) ? rot[(k + 1) & (KP - 1)] : rot[k]; // v[(t+1)%T]
      const float vm = fmaxf(v[k], 1.0f);
      v[k] = v[k] + (num / (vm * dphi)) * (vn - v[k]);
    }
  }

  // ---- write v_out; build scaled series ----
  float Y[KP];
#pragma unroll
  for (int k = 0; k < KP; ++k) {
    const int t = k * 32 + l;
    v_out[b * TLEN + t] = v[k];
    Y[k] = (v[k] - 200.0f) / 1000.0f;
    Ylds[t] = Y[k];
  }

  // ---- means, MAE/RMSE partial sums ----
  float sx = 0.0f, sy = 0.0f, sabs = 0.0f, ssq = 0.0f;
#pragma unroll
  for (int k = 0; k < KP; ++k) { sx += X[k]; sy += Y[k]; }
  sx = waveRedSum(sx); sy = waveRedSum(sy);
  const float mx = sx / (float)TLEN, my = sy / (float)TLEN;
#pragma unroll
  for (int k = 0; k < KP; ++k) {
    const float d = Y[k] - X[k];
    sabs += fabsf(d);
    ssq  += d * d;
  }
  sabs = waveRedSum(sabs); ssq = waveRedSum(ssq);

  // centered series to LDS for Gram marshaling
#pragma unroll
  for (int k = 0; k < KP; ++k) {
    const int t = k * 32 + l;
    pcL[t] = Y[k] - my;
    tcL[t] = X[k] - mx;
  }
  __syncthreads(); // single wave: cheap; orders LDS writes vs reads

  // ---- Pearson Gram matrix G = M^T M, M = [pc | tc] (512x2) ----
  float spc2, spt, stc2;
#if __has_builtin(__builtin_amdgcn_wmma_f32_16x16x4_f32)
  {
    // A (16x4): lanes 0-15 hold M=lane, VGPR v = K=v; lanes 16-31 K=2+v.
    // B (4x16): lanes 0-15 hold N=lane, VGPR v = K=v; lanes 16-31 K=2+v.
    // For the Gram, A and B register images coincide.
    v8f acc = {};
    const int row = l & 15;     // matrix row (A) / col (B)
    const int hi  = l >> 4;     // selects K pair {0,1} vs {2,3}
#pragma unroll 4
    for (int c = 0; c < TLEN / 4; ++c) {
      const int base = c * 4 + 2 * hi;
      const float e0 = (row == 0) ? pcL[base]     : ((row == 1) ? tcL[base]     : 0.0f);
      const float e1 = (row == 0) ? pcL[base + 1] : ((row == 1) ? tcL[base + 1] : 0.0f);
      v2f a; a.x = e0; a.y = e1;
      acc = __builtin_amdgcn_wmma_f32_16x16x4_f32(
          false, a, false, a, (short)0, acc, false, false);
    }
    spc2 = __shfl(acc[0], 0, 32);  // C[0][0] = sum pc*pc
    spt  = __shfl(acc[0], 1, 32);  // C[0][1] = sum pc*tc
    stc2 = __shfl(acc[1], 1, 32);  // C[1][1] = sum tc*tc
  }
#else
  {
    float a2 = 0.0f, b2 = 0.0f, ab = 0.0f;
#pragma unroll
    for (int k = 0; k < KP; ++k) {
      const float pc = Y[k] - my, tc = X[k] - mx;
      a2 += pc * pc; b2 += tc * tc; ab += pc * tc;
    }
    spc2 = waveRedSum(a2); stc2 = waveRedSum(b2); spt = waveRedSum(ab);
  }
#endif
  const float corr = spt / (sqrtf(spc2) * sqrtf(stc2));

  if (l == 0) {
    ws[b * 4 + 0] = sabs;
    ws[b * 4 + 1] = ssq;
    ws[b * 4 + 2] = corr;
  }

  // ---- Soft-DTW wavefront: 1022 serial anti-diagonals, all in VGPRs ----
  // cell i = k*32 + l holds row index i; X[i] in regs, Y[j] from LDS.
  float r1[KP], r2[KP];
  {
    const float x0 = __shfl(X[0], 0, 32);
    float d00 = x0 - Ylds[0];
    d00 *= d00;
#pragma unroll
    for (int k = 0; k < KP; ++k) { r1[k] = BIGF; r2[k] = BIGF; }
    r1[0] = (l == 0) ? d00 : BIGF;   // r0[:,0] = D[0,0]
  }

#pragma unroll 1
  for (int p = 1; p <= 2 * TLEN - 2; ++p) {
    float u_rot[KP], g_rot[KP], rnew[KP];
#pragma unroll
    for (int k = 0; k < KP; ++k) {
      u_rot[k] = __shfl(r1[k], (l + 31) & 31, 32); // lane l gets lane l-1 (wrap)
      g_rot[k] = __shfl(r2[k], (l + 31) & 31, 32);
    }
#pragma unroll
    for (int k = 0; k < KP; ++k) {
      const int i = k * 32 + l;
      const int j = p - i;
      const bool valid = (j >= 0) && (j < TLEN);
      const int jj = min(max(j, 0), TLEN - 1);
      float dd = X[k] - Ylds[jj];
      dd *= dd;
      const float up   = l ? u_rot[k] : (k ? u_rot[k - 1] : BIGF); // R[i-1, j]
      const float diag = l ? g_rot[k] : (k ? g_rot[k - 1] : BIGF); // R[i-1, j-1]
      const float left = r1[k];                                    // R[i,   j-1]
      // soft-min with max-stabilized logsumexp (matches jax semantics)
      const float mn = fminf(up, fminf(left, diag));
      const float s  = expf((mn - up) * 10.0f) +
                       expf((mn - left) * 10.0f) +
                       expf((mn - diag) * 10.0f);
      const float sm = mn - GAMMAF * logf(s);
      rnew[k] = valid ? (dd + sm) : BIGF;
    }
#pragma unroll
    for (int k = 0; k < KP; ++k) { r2[k] = r1[k]; r1[k] = rnew[k]; }
  }

  const float dtw_raw = __shfl(r1[KP - 1], 31, 32); // R[T-1, T-1]
  if (l == 0) ws[b * 4 + 3] = dtw_raw;
}

__global__ __launch_bounds__(32) void finalize_kernel(
    const float* __restrict__ ws, float* __restrict__ out, int B)
{
  const int l = threadIdx.x;
  float sabs = 0.0f, ssq = 0.0f, corr = 0.0f, adtw = 0.0f;
  if (l < B) {
    sabs = ws[l * 4 + 0];
    ssq  = ws[l * 4 + 1];
    corr = ws[l * 4 + 2];
    adtw = fabsf(ws[l * 4 + 3]);
  }
  sabs = waveRedSum(sabs); ssq = waveRedSum(ssq);
  corr = waveRedSum(corr); adtw = waveRedSum(adtw);
  if (l == 0) {
    out[0] = sabs;                    // mae_loss  = sum |diff|
    out[1] = 1.0f - corr / (float)B;  // pcc_loss
    out[2] = adtw / (float)B;         // dtw_loss  = mean |dtw_raw|
    out[3] = sqrtf(ssq);              // rmse_loss = sqrt(sum diff^2)
  }
}

extern "C" void kernel_launch(void* const* d_in, const int* in_sizes, int n_in,
                              void* d_out, int out_size, void* d_ws, size_t ws_size,
                              hipStream_t stream) {
  const float* pred_map = (const float*)d_in[0];
  const int*   lat_idx  = (const int*)d_in[1];
  const float* omega    = (const float*)d_in[2];
  const float* omni_ts  = (const float*)d_in[3];
  float* out = (float*)d_out;
  float* ws  = (float*)d_ws;

  const int B = in_sizes[2];                 // omega: [B]
  const int W = in_sizes[0] / (B * TLEN);    // pred_map: [B,1,T,W]

  hipLaunchKernelGGL(hux_dtw_kernel, dim3(B), dim3(32), 0, stream,
                     pred_map, lat_idx, omega, omni_ts, out, ws, W);
  hipLaunchKernelGGL(finalize_kernel, dim3(1), dim3(32), 0, stream,
                     ws, out + B * TLEN, B);
}